// StateOuterProductCumsum_44238163148997
// MI455X (gfx1250) — compile-verified
//
#include <hip/hip_runtime.h>
#include <hip/hip_bf16.h>

// Problem constants (fixed by reference: T=1024, B=8, d=64, n=64)
#define T   1024
#define B   8
#define D   64
#define N   64
#define TC  32            // chunk length along T
#define C   (T / TC)      // 32 chunks
#define EPSF 1e-8f

typedef float v2f __attribute__((ext_vector_type(2)));
typedef float v4f __attribute__((ext_vector_type(4)));
typedef float v8f __attribute__((ext_vector_type(8)));

// ---------------------------------------------------------------------------
// Pass 1: per (b,n) sequential scan over T.
//   kdec[t,b,n] = k[t,b,n] * cumprod[t,b,n] / (cumprod[T-1,b,n] + eps)
// 512 independent chains; loads/stores coalesced across (b,n) at each t.
// ---------------------------------------------------------------------------
__global__ __launch_bounds__(256) void k_decay(const float* __restrict__ alpha,
                                               const float* __restrict__ kk,
                                               float* __restrict__ kdec) {
    int idx = blockIdx.x * blockDim.x + threadIdx.x;   // 0 .. B*N-1
    if (idx >= B * N) return;
    float runlog = 0.0f;
    for (int t = 0; t < T; ++t) {
        float a = alpha[(size_t)t * (B * N) + idx];
        a = fmaxf(a, EPSF);
        runlog += logf(a);
        kdec[(size_t)t * (B * N) + idx] = expf(runlog);   // stage cumprod
    }
    float inv = 1.0f / (expf(runlog) + EPSF);
    for (int t = 0; t < T; ++t) {
        size_t o = (size_t)t * (B * N) + idx;
        kdec[o] = kdec[o] * inv * kk[o];
    }
}

// ---------------------------------------------------------------------------
// Pass 2: chunk partial sums via fp32 WMMA.
//   M_c[d,n] = sum_{t in chunk c} v[t,b,d] * kdec[t,b,n]
// One block per (c,b): a 64x64x32 GEMM, A[d,t]=v[t,d] (16x4 frags),
// B[t,n]=kdec[t,n]. 8 waves; wave w owns m-tile w>>1 and two n-tiles.
// f32 16x16x4 fragment layout (ISA 7.12.2): lanes 0-15 carry rows, the
// lane-half selects the K pair held in the two A/B VGPRs.
// ---------------------------------------------------------------------------
__global__ __launch_bounds__(256) void k_chunk_gemm(const float* __restrict__ v,
                                                    const float* __restrict__ kdec,
                                                    float* __restrict__ M) {
    const int c = blockIdx.x / B;
    const int b = blockIdx.x % B;
    const int tid   = threadIdx.x;
    const int wave  = tid >> 5;
    const int lane  = tid & 31;
    const int mt    = wave >> 1;              // m-tile 0..3
    const int ntb   = (wave & 1) * 2;         // n-tiles ntb, ntb+1
    const int lhalf = lane >> 4;              // selects K pair (0->K0,1 ; 1->K2,3)
    const int l16   = lane & 15;

    v8f acc0 = {};
    v8f acc1 = {};
    const int t0 = c * TC;

    #pragma unroll
    for (int ks = 0; ks < TC / 4; ++ks) {
        const int tk = t0 + ks * 4 + lhalf * 2;
        // A fragment: A[M=d, K=t]; lane row = d = mt*16 + l16
        const int d = mt * 16 + l16;
        v2f a;
        a.x = v[((size_t)(tk    ) * B + b) * D + d];
        a.y = v[((size_t)(tk + 1) * B + b) * D + d];
        // B fragments: B[K=t, N=n]; lane col = n
        const float* kr0 = kdec + ((size_t)(tk    ) * B + b) * N;
        const float* kr1 = kdec + ((size_t)(tk + 1) * B + b) * N;
        const int n0 = ntb * 16 + l16;
        const int n1 = (ntb + 1) * 16 + l16;
        v2f b0, b1;
        b0.x = kr0[n0]; b0.y = kr1[n0];
        b1.x = kr0[n1]; b1.y = kr1[n1];
        acc0 = __builtin_amdgcn_wmma_f32_16x16x4_f32(false, a, false, b0,
                                                     (short)0, acc0, false, false);
        acc1 = __builtin_amdgcn_wmma_f32_16x16x4_f32(false, a, false, b1,
                                                     (short)0, acc1, false, false);
    }

    // Store C/D: lane<16 -> rows m = mt*16 + r ; lane>=16 -> m = mt*16 + 8 + r
    float* Mc = M + ((size_t)c * B + b) * (D * N);
    const int mbase = mt * 16 + lhalf * 8;
    const int nn0 = ntb * 16 + l16;
    const int nn1 = (ntb + 1) * 16 + l16;
    #pragma unroll
    for (int r = 0; r < 8; ++r) {
        Mc[(size_t)(mbase + r) * N + nn0] = acc0[r];
        Mc[(size_t)(mbase + r) * N + nn1] = acc1[r];
    }
}

// ---------------------------------------------------------------------------
// Pass 2b: in-place exclusive scan of the C chunk matrices (per b,d,n element).
// ---------------------------------------------------------------------------
__global__ __launch_bounds__(256) void k_chunk_scan(float* __restrict__ M) {
    const int idx = blockIdx.x * blockDim.x + threadIdx.x;   // 0 .. B*D*N-1
    float run = 0.0f;
    #pragma unroll 1
    for (int c = 0; c < C; ++c) {
        const size_t o = (size_t)c * (B * D * N) + idx;
        const float m = M[o];
        M[o] = run;
        run += m;
    }
}

// ---------------------------------------------------------------------------
// Pass 3: bandwidth pass. One block per (c,b); thread owns a 4(d) x 4(n)
// accumulator tile seeded from the chunk base, scans 32 t-steps, and emits
// 128-bit non-temporal stores (128 MiB total stream).
// ---------------------------------------------------------------------------
__global__ __launch_bounds__(256) void k_scan_store(const float* __restrict__ v,
                                                    const float* __restrict__ kdec,
                                                    const float* __restrict__ O,
                                                    float* __restrict__ S) {
    const int c = blockIdx.x / B;
    const int b = blockIdx.x % B;
    const int tid = threadIdx.x;
    const int nq = tid & 15;          // n quad index -> n0 = nq*4
    const int dq = tid >> 4;          // 0..15        -> d  = dq*4 + i
    const int n0 = nq * 4;

    v4f acc[4];
    const float* Ob = O + ((size_t)c * B + b) * (D * N);
    #pragma unroll
    for (int i = 0; i < 4; ++i)
        acc[i] = *(const v4f*)(Ob + (size_t)(dq * 4 + i) * N + n0);

    const int tEnd = c * TC + TC;
    #pragma unroll 1
    for (int t = c * TC; t < tEnd; ++t) {
        const v4f kv = *(const v4f*)(kdec + ((size_t)t * B + b) * N + n0);
        const float* vr = v + ((size_t)t * B + b) * D + dq * 4;
        float* St = S + ((size_t)t * B + b) * (size_t)(D * N);
        #pragma unroll
        for (int i = 0; i < 4; ++i) {
            const float vi = vr[i];
            acc[i].x = fmaf(vi, kv.x, acc[i].x);
            acc[i].y = fmaf(vi, kv.y, acc[i].y);
            acc[i].z = fmaf(vi, kv.z, acc[i].z);
            acc[i].w = fmaf(vi, kv.w, acc[i].w);
            __builtin_nontemporal_store(acc[i],
                (v4f*)(St + (size_t)(dq * 4 + i) * N + n0));
        }
    }
}

// ---------------------------------------------------------------------------
// Host launcher. Inputs: d_in[0]=v (T,B,D), d_in[1]=k (T,B,N),
// d_in[2]=alpha (T,B,N). Output: S (T,B,D,N) fp32.
// Workspace layout: [0, 2MiB) kdec ; [2MiB, 6MiB) chunk matrices M/O.
// ---------------------------------------------------------------------------
extern "C" void kernel_launch(void* const* d_in, const int* in_sizes, int n_in,
                              void* d_out, int out_size, void* d_ws, size_t ws_size,
                              hipStream_t stream) {
    const float* v     = (const float*)d_in[0];
    const float* kk    = (const float*)d_in[1];
    const float* alpha = (const float*)d_in[2];
    float* S = (float*)d_out;

    float* kdec = (float*)d_ws;                                  // T*B*N floats
    float* M    = (float*)((char*)d_ws + (size_t)T * B * N * 4); // C*B*D*N floats

    // Pass 1: decay * k  (512 chains)
    k_decay<<<(B * N + 255) / 256, 256, 0, stream>>>(alpha, kk, kdec);

    // Pass 2: per-chunk 64x64x32 GEMMs via WMMA f32 16x16x4
    k_chunk_gemm<<<C * B, 256, 0, stream>>>(v, kdec, M);

    // Pass 2b: exclusive scan over chunks (in place)
    k_chunk_scan<<<(B * D * N) / 256, 256, 0, stream>>>(M);

    // Pass 3: within-chunk scan + streaming NT stores
    k_scan_store<<<C * B, 256, 0, stream>>>(v, kdec, M, S);
}